// FixedPairThresholdMerge_32220844655197
// MI455X (gfx1250) — compile-verified
//
#include <hip/hip_runtime.h>
#include <math.h>

typedef __attribute__((ext_vector_type(2))) float v2f;
typedef __attribute__((ext_vector_type(4))) float v4f;
typedef __attribute__((ext_vector_type(8))) float v8f;

#define B_ 16
#define T_ 4096
#define C_ 1024
#define H_ 64
#define P_ (T_ / 2)   // 2048 pairs per batch

// ---------------------------------------------------------------------------
// Kernel A: fused pair cosine-similarity + deterministic column-sum partials.
// Grid: B_*64 blocks of 256 threads (8 waves). Each wave handles 4 pairs.
// Each block covers 32 pairs (64 rows) of one batch; column sums accumulate in
// registers (8 x float4 per lane), staged through LDS, combined in fixed order.
// ---------------------------------------------------------------------------
__global__ __launch_bounds__(256) void pair_stats_kernel(
    const float* __restrict__ metric, float* __restrict__ sim,
    float* __restrict__ gpart) {
  __shared__ float lds_col[8 * C_];  // 32 KB
  const int tid  = threadIdx.x;
  const int wave = tid >> 5;
  const int lane = tid & 31;
  const int b    = blockIdx.x >> 6;  // 64 blocks per batch
  const int blk  = blockIdx.x & 63;

  v4f col[8] = {};

  for (int pp = 0; pp < 4; ++pp) {
    const int p = blk * 32 + wave * 4 + pp;
    const float* r0 = metric + ((size_t)b * T_ + 2 * (size_t)p) * C_;
    const float* r1 = r0 + C_;
    float dot = 0.f, n0 = 0.f, n1 = 0.f;
#pragma unroll
    for (int k = 0; k < 8; ++k) {
      const int c = k * 128 + lane * 4;  // 512B-contiguous per wave access
      v4f a = *(const v4f*)(r0 + c);
      v4f d = *(const v4f*)(r1 + c);
      dot += a.x * d.x + a.y * d.y + a.z * d.z + a.w * d.w;
      n0  += a.x * a.x + a.y * a.y + a.z * a.z + a.w * a.w;
      n1  += d.x * d.x + d.y * d.y + d.z * d.z + d.w * d.w;
      col[k] += a + d;
    }
#pragma unroll
    for (int off = 16; off > 0; off >>= 1) {
      dot += __shfl_xor(dot, off, 32);
      n0  += __shfl_xor(n0,  off, 32);
      n1  += __shfl_xor(n1,  off, 32);
    }
    if (lane == 0) {
      sim[b * P_ + p] =
          dot / (fmaxf(sqrtf(n0), 1e-12f) * fmaxf(sqrtf(n1), 1e-12f));
    }
  }
  // Stage per-wave column sums, combine in fixed wave order (deterministic).
#pragma unroll
  for (int k = 0; k < 8; ++k)
    *(v4f*)(&lds_col[wave * C_ + k * 128 + lane * 4]) = col[k];
  __syncthreads();
  v4f tot = {};
#pragma unroll
  for (int w = 0; w < 8; ++w) tot += *(const v4f*)(&lds_col[w * C_ + tid * 4]);
  *(v4f*)(&gpart[((size_t)(b * 64 + blk)) * C_ + tid * 4]) = tot;
}

// ---------------------------------------------------------------------------
// Kernel A2: fixed-order reduction of 64 block partials -> gsum (B x C).
// ---------------------------------------------------------------------------
__global__ __launch_bounds__(256) void gsum_reduce_kernel(
    const float* __restrict__ gpart, float* __restrict__ gsum) {
  const int idx = blockIdx.x * 256 + threadIdx.x;  // 0 .. B_*C_-1
  const int b = idx >> 10;
  const int c = idx & (C_ - 1);
  float s = 0.f;
  for (int j = 0; j < 64; ++j) s += gpart[((size_t)(b * 64 + j)) * C_ + c];
  gsum[idx] = s;
}

// ---------------------------------------------------------------------------
// Kernel B: gate MLP via V_WMMA_F32_16X16X4_F32 (full-precision f32 WMMA).
// One block, 128 threads (4 waves). M=16 (=B), N tile = wave*16, K loops 1024.
// A frag (16x4 f32): lane%16 = M, lane/16 selects K pair, vgpr j -> K within.
// B frag (4x16 f32): mirrored, lane%16 = N.
// D tile (16x16 f32): vgpr r, lanes 0-15 -> M=r, lanes 16-31 -> M=r+8, N=lane%16.
// ---------------------------------------------------------------------------
__global__ __launch_bounds__(128) void gate_mlp_wmma_kernel(
    const float* __restrict__ gsum, const float* __restrict__ w1,
    const float* __restrict__ b1, const float* __restrict__ w2,
    const float* __restrict__ b2, float* __restrict__ theta) {
  __shared__ float lds_hw[B_ * H_];  // 16 x 64 products, fixed-order reduce
  const int tid   = threadIdx.x;
  const int wave  = tid >> 5;            // N tile 0..3
  const int lane  = tid & 31;
  const int M     = lane & 15;
  const int koff  = (lane >> 4) << 1;    // 0 or 2
  const int ncol  = (lane & 15) + wave * 16;
  const float invT = 1.0f / (float)T_;

  v8f acc = {};
  for (int kk = 0; kk < C_; kk += 4) {
    v2f a, bb;
    a.x  = gsum[M * C_ + kk + koff] * invT;
    a.y  = gsum[M * C_ + kk + koff + 1] * invT;
    bb.x = w1[(kk + koff) * H_ + ncol];
    bb.y = w1[(kk + koff + 1) * H_ + ncol];
    acc = __builtin_amdgcn_wmma_f32_16x16x4_f32(
        /*neg_a=*/false, a, /*neg_b=*/false, bb,
        /*c_mod=*/(short)0, acc, /*reuse_a=*/false, /*reuse_b=*/false);
  }
  const float bias  = b1[ncol];
  const float wv    = w2[ncol];
  const int   mbase = (lane >> 4) << 3;
#pragma unroll
  for (int r = 0; r < 8; ++r) {
    float hpre = acc[r] + bias;
    // exact GELU: x * 0.5 * (1 + erf(x / sqrt(2)))
    float h = 0.5f * hpre * (1.0f + erff(hpre * 0.70710678f));
    lds_hw[(mbase + r) * H_ + ncol] = h * wv;
  }
  __syncthreads();
  if (tid < B_) {
    float s = b2[0];
    for (int n = 0; n < H_; ++n) s += lds_hw[tid * H_ + n];
    float t = 1.0f / (1.0f + expf(-s));
    theta[tid] = -0.5f + t;  // THETA_MIN + (THETA_MAX-THETA_MIN)*t
  }
}

// ---------------------------------------------------------------------------
// Kernel C: per-batch inclusive scan of keep flags (keep = sim < theta).
// One block of 256 threads per batch; thread handles 8 pairs.
// ---------------------------------------------------------------------------
__global__ __launch_bounds__(256) void scan_kernel(
    const float* __restrict__ sim, const float* __restrict__ theta,
    int* __restrict__ kc, int* __restrict__ nkeep) {
  __shared__ int lds_cnt[256];
  __shared__ int lds_total;
  const int b = blockIdx.x;
  const int t = threadIdx.x;
  const float th = theta[b];
  int flags = 0, cnt = 0;
#pragma unroll
  for (int j = 0; j < 8; ++j) {
    const int k = (sim[b * P_ + t * 8 + j] < th) ? 1 : 0;
    flags |= (k << j);
    cnt += k;
  }
  lds_cnt[t] = cnt;
  __syncthreads();
  if (t == 0) {
    int run = 0;
    for (int j = 0; j < 256; ++j) { int e = lds_cnt[j]; lds_cnt[j] = run; run += e; }
    lds_total = run;
  }
  __syncthreads();
  int run = lds_cnt[t];
#pragma unroll
  for (int j = 0; j < 8; ++j) {
    run += (flags >> j) & 1;
    kc[b * P_ + t * 8 + j] = run;
  }
  if (t == 0) nkeep[b] = lds_total;
}

// ---------------------------------------------------------------------------
// Kernel D: scatter. One block (256 threads) per pair; float4 per thread covers
// C=1024. Every pair writes exactly 2 output rows (keep: src+dst; merge:
// avg+zero), which tiles out[0..T) completely -> no separate clear pass.
// Nontemporal: streaming 512 MB, larger than 192 MB L2.
// ---------------------------------------------------------------------------
__global__ __launch_bounds__(256) void scatter_kernel(
    const float* __restrict__ x, const float* __restrict__ sim,
    const float* __restrict__ theta, const int* __restrict__ kc,
    const int* __restrict__ nkeep, float* __restrict__ out) {
  const int gp = blockIdx.x;
  const int b  = gp >> 11;         // / 2048
  const int p  = gp & (P_ - 1);
  const int c  = threadIdx.x * 4;
  const bool keep = sim[b * P_ + p] < theta[b];
  const int  kcv  = kc[b * P_ + p];
  const int  nk   = nkeep[b];
  const float* src = x + ((size_t)b * T_ + 2 * (size_t)p) * C_;
  v4f s = __builtin_nontemporal_load((const v4f*)(src + c));
  v4f d = __builtin_nontemporal_load((const v4f*)(src + C_ + c));
  int rowA, rowB;
  v4f outA, outB;
  if (keep) {
    rowA = kcv - 1;       outA = s;
    rowB = P_ + kcv - 1;  outB = d;
  } else {
    const int mc = (p + 1) - kcv;       // inclusive merge count
    rowA = nk + mc - 1;       outA = (s + d) * 0.5f;
    rowB = P_ + nk + mc - 1;  outB = (v4f){};   // tail rows must be zero
  }
  float* ob = out + (size_t)b * T_ * C_;
  __builtin_nontemporal_store(outA, (v4f*)(ob + (size_t)rowA * C_ + c));
  __builtin_nontemporal_store(outB, (v4f*)(ob + (size_t)rowB * C_ + c));
}

// ---------------------------------------------------------------------------
extern "C" void kernel_launch(void* const* d_in, const int* in_sizes, int n_in,
                              void* d_out, int out_size, void* d_ws,
                              size_t ws_size, hipStream_t stream) {
  const float* metric = (const float*)d_in[0];
  const float* x      = (const float*)d_in[1];
  const float* w1     = (const float*)d_in[2];
  const float* b1     = (const float*)d_in[3];
  const float* w2     = (const float*)d_in[4];
  const float* b2     = (const float*)d_in[5];
  float* out = (float*)d_out;

  char* ws = (char*)d_ws;
  size_t off = 0;
  auto take = [&](size_t bytes) {
    void* p = ws + off;
    off = (off + bytes + 255) & ~(size_t)255;
    return p;
  };
  float* sim   = (float*)take((size_t)B_ * P_ * sizeof(float));       // 128 KB
  float* gpart = (float*)take((size_t)B_ * 64 * C_ * sizeof(float));  //   4 MB
  float* gsum  = (float*)take((size_t)B_ * C_ * sizeof(float));       //  64 KB
  float* theta = (float*)take((size_t)B_ * sizeof(float));
  int*   kc    = (int*)take((size_t)B_ * P_ * sizeof(int));           // 128 KB
  int*   nkeep = (int*)take((size_t)B_ * sizeof(int));
  (void)ws_size; (void)in_sizes; (void)n_in; (void)out_size;

  pair_stats_kernel<<<B_ * 64, 256, 0, stream>>>(metric, sim, gpart);
  gsum_reduce_kernel<<<(B_ * C_) / 256, 256, 0, stream>>>(gpart, gsum);
  gate_mlp_wmma_kernel<<<1, 128, 0, stream>>>(gsum, w1, b1, w2, b2, theta);
  scan_kernel<<<B_, 256, 0, stream>>>(sim, theta, kc, nkeep);
  scatter_kernel<<<B_ * P_, 256, 0, stream>>>(x, sim, theta, kc, nkeep, out);
}